// CRF_47785806135961
// MI455X (gfx1250) — compile-verified
//
#include <hip/hip_runtime.h>
#include <hip/hip_bf16.h>
#include <stdint.h>

// ---- problem constants (match reference) ----
#define TAGS   52      // T = TAGSET + 2
#define SSTART 50
#define SSTOP  51
#define SLEN   512     // S
#define TPAD   64      // padded tag dimension (2 slots per lane, wave32)
#define NEGV   (-10000.0f)

#define WAVES_PER_BLOCK 4
#define BLOCK_THREADS   (WAVES_PER_BLOCK * 32)

// ---- LDS layout (dynamic) ----
#define TRANS_BYTES (TAGS * TPAD * 4)                       // 13312: trans[52][64]
#define BP_BYTES    (SLEN * TPAD)                           // 32768: u8 bp, row stride 64
#define FBUF_BYTES  (2 * TPAD * 4)                          // 512: double-buffered feat row
#define PER_WAVE    (BP_BYTES + FBUF_BYTES)                 // 33280
#define SMEM_BYTES  (TRANS_BYTES + WAVES_PER_BLOCK * PER_WAVE)  // 146432 -> 2 blocks/WGP

__global__ void __launch_bounds__(BLOCK_THREADS)
crf_viterbi_kernel(const float* __restrict__ feats,      // [B,S,T]
                   const int*   __restrict__ mask,       // [B,S] prefix mask (0/1)
                   const float* __restrict__ trans,      // [T,T]
                   int*         __restrict__ out,        // [B,S] int32
                   int B)
{
    extern __shared__ char smem[];
    const int tid  = threadIdx.x;
    const int lane = tid & 31;
    const int wv   = tid >> 5;
    const int b    = blockIdx.x * WAVES_PER_BLOCK + wv;

    // ---- cooperative load of transitions into LDS: [52][64] padded ----
    // 52*52 = 2704 elements; flat index strided by 128, row/col split by a div per element.
    float* transLds = (float*)smem;
    {
        int r0 = tid / TAGS;            // small const-divisor div, prologue only
        int c0 = tid - r0 * TAGS;
        // BLOCK_THREADS=128: stride pattern +128 over flat index
        for (int idx = tid; idx < TAGS * TAGS; idx += BLOCK_THREADS) {
            int r = idx / TAGS;
            int c = idx - r * TAGS;
            transLds[r * TPAD + c] = trans[idx];
        }
        (void)r0; (void)c0;
    }
    __syncthreads();

    if (b >= B) return;

    const uint32_t ldsbase = __builtin_amdgcn_groupstaticsize();
    const uint32_t bpOff   = (uint32_t)TRANS_BYTES + (uint32_t)wv * PER_WAVE;
    const uint32_t fOff    = bpOff + BP_BYTES;
    uint8_t* bpLds = (uint8_t*)(smem + bpOff);
    float*   fb    = (float*)(smem + fOff);   // fb[buf*64 + tag], buf in {0,1}

    // ---- sequence length = popcount of prefix mask (wave reduction) ----
    const int* mrow = mask + (size_t)b * SLEN;
    int len = 0;
    for (int i = lane; i < SLEN; i += 32) len += mrow[i];
    #pragma unroll
    for (int d = 16; d; d >>= 1) len += __shfl_xor(len, d, 32);
    if (len < 1) len = 1;

    const float* frow = feats + (size_t)b * SLEN * TAGS;

    // ---- async feat-row loader: 208B = 13 lanes x b128 into LDS buffer ----
    auto issue = [&](int t, int buf) {
        if (lane < 13) {
            uint32_t la = ldsbase + fOff + (uint32_t)buf * 256u + (uint32_t)lane * 16u;
            const float* ga = frow + (size_t)t * TAGS + lane * 4;
            asm volatile("global_load_async_to_lds_b128 %0, %1, off"
                         :: "v"(la), "v"(ga) : "memory");
        }
    };

    // prefetch step t=1 into buffer 1 while we set up alpha0
    if (len > 1) issue(1, 1);

    // ---- alpha0 = trans[START][cur] + constrained feats[b,0,cur] ----
    const int  cur0  = lane;                    // 0..31
    const int  cur1  = lane + 32;               // 32..63 (52..63 are pads)
    const bool cons1 = (cur1 == SSTART) || (cur1 == SSTOP);
    float f00 = frow[cur0];                                   // cur0 < 52 always
    float f01 = (cur1 < TAGS) ? frow[cur1] : 0.0f;
    if (cons1) f01 = NEGV;                                    // emission constraint
    float aLo = transLds[SSTART * TPAD + cur0] + f00;
    float aHi = (cur1 < TAGS) ? (transLds[SSTART * TPAD + cur1] + f01) : -3.0e38f;

    // ---- Viterbi recurrence over t = 1 .. len-1 ----
    for (int t = 1; t < len; ++t) {
        const int buf = t & 1;
        if (t + 1 < len) {
            issue(t + 1, (t + 1) & 1);                        // prefetch next row
            asm volatile("s_wait_asynccnt 0x1" ::: "memory"); // row t resident
        } else {
            asm volatile("s_wait_asynccnt 0x0" ::: "memory"); // last row resident
        }

        float ft0 = fb[buf * 64 + lane];
        float ft1 = fb[buf * 64 + lane + 32];
        if (cons1) ft1 = NEGV;                                // lanes 18,19 slot-hi

        float best0 = -3.0e38f, best1 = -3.0e38f;
        int   bp0 = 0, bp1 = 0;

        // fully unrolled: immediate readlane selects + immediate DS offsets
        #pragma unroll
        for (int p = 0; p < 32; ++p) {
            float a = __builtin_bit_cast(float,
                        __builtin_amdgcn_readlane(__builtin_bit_cast(int, aLo), p));
            float s0 = a + transLds[p * TPAD + cur0];
            float s1 = a + transLds[p * TPAD + cur1];
            if (s0 > best0) { best0 = s0; bp0 = p; }
            if (s1 > best1) { best1 = s1; bp1 = p; }
        }
        #pragma unroll
        for (int p = 0; p < TAGS - 32; ++p) {                 // prev = 32..51
            float a = __builtin_bit_cast(float,
                        __builtin_amdgcn_readlane(__builtin_bit_cast(int, aHi), p));
            const int pv = p + 32;
            float s0 = a + transLds[pv * TPAD + cur0];
            float s1 = a + transLds[pv * TPAD + cur1];
            if (s0 > best0) { best0 = s0; bp0 = pv; }
            if (s1 > best1) { best1 = s1; bp1 = pv; }
        }

        aLo = best0 + ft0;
        aHi = best1 + ft1;
        // pack backpointers: byte 2*lane = slot cur0, byte 2*lane+1 = slot cur1
        *(uint16_t*)(bpLds + t * TPAD + lane * 2) = (uint16_t)(bp0 | (bp1 << 8));
    }
    asm volatile("s_wait_asynccnt 0x0" ::: "memory");         // drain any prefetch

    // ---- best_last = argmax_cur(alpha[cur] + trans[cur][STOP]) ----
    float c0 = aLo + transLds[cur0 * TPAD + SSTOP];
    float c1 = (cur1 < TAGS) ? (aHi + transLds[cur1 * TPAD + SSTOP]) : -3.0e38f;
    float bv; int bi;
    if (c1 > c0) { bv = c1; bi = cur1; } else { bv = c0; bi = cur0; }
    #pragma unroll
    for (int d = 16; d; d >>= 1) {
        float ov = __shfl_xor(bv, d, 32);
        int   oi = __shfl_xor(bi, d, 32);
        if (ov > bv || (ov == bv && oi < bi)) { bv = ov; bi = oi; }  // ties -> lowest idx
    }

    // ---- traceback (lane 0, all backpointers in LDS) ----
    int* orow = out + (size_t)b * SLEN;
    if (lane == 0) {
        int ptr = bi;
        orow[len - 1] = ptr;
        for (int pos = len - 1; pos >= 1; --pos) {
            ptr = bpLds[pos * TPAD + 2 * (ptr & 31) + (ptr >> 5)];
            orow[pos - 1] = ptr;
        }
    }
    // positions past length-1 are 0
    for (int p2 = len + lane; p2 < SLEN; p2 += 32) orow[p2] = 0;
}

extern "C" void kernel_launch(void* const* d_in, const int* in_sizes, int n_in,
                              void* d_out, int out_size, void* d_ws, size_t ws_size,
                              hipStream_t stream) {
    const float* feats = (const float*)d_in[0];   // [B,S,T] f32
    const int*   mask  = (const int*)d_in[1];     // [B,S]
    // d_in[2] = tags: unused by Viterbi decode
    const float* trans = (const float*)d_in[3];   // [T,T] f32
    int* out = (int*)d_out;                       // [B,S] int32

    const int B = in_sizes[1] / SLEN;
    const int blocks = (B + WAVES_PER_BLOCK - 1) / WAVES_PER_BLOCK;
    hipLaunchKernelGGL(crf_viterbi_kernel, dim3(blocks), dim3(BLOCK_THREADS),
                       SMEM_BYTES, stream, feats, mask, trans, out, B);
}